// BilinearLinformerCapsuleFC_26688926777730
// MI455X (gfx1250) — compile-verified
//
#include <hip/hip_runtime.h>
#include <hip/hip_bf16.h>
#include <math.h>

// ---------------- problem constants ----------------
#define BB     32
#define IN_N   32
#define IN_D   16
#define HH     14
#define OUT_N  32
#define OUT_D  16
#define HO     7
#define HID    64
#define LN_EPS 1e-5f

#define JROWS  1568           // OUT_N*HO*HO
#define JTILES 98             // 1568/16
#define NROWS  2097           // IN_N*HID + HO*HO
#define NPAD   2112           // padded to 132*16
#define NTILES 132

typedef __attribute__((ext_vector_type(2))) float v2f;
typedef __attribute__((ext_vector_type(8))) float v8f;

// f32 WMMA: D(16x16) = A(16x4) * B(4x16) + C   (chain 4x for K=16)
__device__ inline v8f wmma_f32_4(v2f a, v2f b, v8f c) {
  return __builtin_amdgcn_wmma_f32_16x16x4_f32(
      /*neg_a=*/false, a, /*neg_b=*/false, b,
      /*c_mod=*/(short)0, c, /*reuse_a=*/false, /*reuse_b=*/false);
}

// ---------------- kernel 1: grouped conv on current_pose -> cpconv[b][c=n*16+dout][p]
__global__ __launch_bounds__(256)
void conv_current_kernel(const float* __restrict__ cur_pose,
                         const float* __restrict__ cur_w,
                         float* __restrict__ cpconv) {
  const int bn = blockIdx.x;            // b*32 + n
  const int n  = bn & 31;
  __shared__ float sIn[HH * HH * IN_D]; // [y][x][din]
  __shared__ float sW[16 * 16 * 9];     // [dout][din][k]
  const float* inp = cur_pose + (size_t)bn * (HH * HH * IN_D);
  const float* wp  = cur_w + (size_t)(n * 16) * (16 * 9);
  for (int t = threadIdx.x; t < HH * HH * IN_D; t += 256) sIn[t] = inp[t];
  for (int t = threadIdx.x; t < 16 * 16 * 9;    t += 256) sW[t]  = wp[t];
  __syncthreads();
  const int p = threadIdx.x;
  if (p < HH * HH) {
    const int y = p / HH, x = p % HH;
    float acc[16];
#pragma unroll
    for (int d = 0; d < 16; ++d) acc[d] = 0.f;
    for (int ky = 0; ky < 3; ++ky) {
      const int yy = y + ky - 1;
      if (yy < 0 || yy >= HH) continue;
      for (int kx = 0; kx < 3; ++kx) {
        const int xx = x + kx - 1;
        if (xx < 0 || xx >= HH) continue;
        const float* iv = &sIn[(yy * HH + xx) * IN_D];
        for (int din = 0; din < 16; ++din) {
          const float v = iv[din];
          const float* wv = &sW[din * 9 + ky * 3 + kx];
#pragma unroll
          for (int dout = 0; dout < 16; ++dout)
            acc[dout] += v * wv[dout * 144];
        }
      }
    }
    float* op = cpconv + (size_t)bn * (16 * HH * HH);
#pragma unroll
    for (int dout = 0; dout < 16; ++dout) op[dout * (HH * HH) + p] = acc[dout];
  }
}

// ---------------- kernel 2: Linformer K projection -> kbuf[b][n*64+h][e]
__global__ __launch_bounds__(256)
void build_k_kernel(const float* __restrict__ cpconv,
                    const float* __restrict__ E_proj,
                    float* __restrict__ kbuf) {
  const int bn = blockIdx.x;
  const int b = bn >> 5, n = bn & 31;
  __shared__ float sC[16 * 196];        // rows e: cpconv[b][e*32+n][p]
  const float* cb = cpconv + (size_t)b * (512 * 196);
  for (int t = threadIdx.x; t < 16 * 196; t += 256) {
    const int e = t / 196, p = t % 196;
    sC[t] = cb[(size_t)(e * 32 + n) * 196 + p];
  }
  __syncthreads();
  const float* Ep = E_proj + (size_t)n * 196 * HID;
#pragma unroll
  for (int u = 0; u < 4; ++u) {
    const int idx = threadIdx.x + 256 * u;   // 0..1023 = 64 h x 16 e
    const int h = idx >> 4, e = idx & 15;
    float acc = 0.f;
    for (int p = 0; p < 196; ++p) acc += sC[e * 196 + p] * Ep[p * HID + h];
    kbuf[((size_t)b * NPAD + n * HID + h) * 16 + e] = acc;
  }
}

// ---------------- kernel 3: rel-embedding rows + zero pad of K
__global__ __launch_bounds__(256)
void fill_rel_kernel(const float* __restrict__ rel, float* __restrict__ kbuf) {
  const int b = blockIdx.x;
#pragma unroll
  for (int u = 0; u < 4; ++u) {
    const int idx = threadIdx.x + 256 * u;   // 64 rows x 16 e
    const int s = idx >> 4, e = idx & 15;
    const float v = (s < HO * HO) ? rel[e * (HO * HO) + s] : 0.f;
    kbuf[((size_t)b * NPAD + 2048 + s) * 16 + e] = v;
  }
}

// ---------------- kernel 4: grouped conv on next_pose -> qbuf[b][c*49+s] (flat == [j][e])
__global__ __launch_bounds__(64)
void conv_q_kernel(const float* __restrict__ next_pose,
                   const float* __restrict__ next_w,
                   float* __restrict__ qbuf) {
  const int bn = blockIdx.x;
  const int b = bn >> 5, n = bn & 31;
  __shared__ float sIn[HO * HO * 16];
  __shared__ float sW[16 * 16 * 9];
  const float* inp = next_pose + (size_t)bn * (HO * HO * 16);
  const float* wp  = next_w + (size_t)(n * 16) * 144;
  for (int t = threadIdx.x; t < HO * HO * 16; t += 64) sIn[t] = inp[t];
  for (int t = threadIdx.x; t < 16 * 144;     t += 64) sW[t]  = wp[t];
  __syncthreads();
  const int p = threadIdx.x;
  if (p < HO * HO) {
    const int y = p / HO, x = p % HO;
    float acc[16];
#pragma unroll
    for (int d = 0; d < 16; ++d) acc[d] = 0.f;
    for (int ky = 0; ky < 3; ++ky) {
      const int yy = y + ky - 1;
      if (yy < 0 || yy >= HO) continue;
      for (int kx = 0; kx < 3; ++kx) {
        const int xx = x + kx - 1;
        if (xx < 0 || xx >= HO) continue;
        const float* iv = &sIn[(yy * HO + xx) * 16];
        for (int din = 0; din < 16; ++din) {
          const float v = iv[din];
          const float* wv = &sW[din * 9 + ky * 3 + kx];
#pragma unroll
          for (int dout = 0; dout < 16; ++dout)
            acc[dout] += v * wv[dout * 144];
        }
      }
    }
    float* op = qbuf + (size_t)b * (JROWS * 16) + (size_t)(n * 16) * (HO * HO);
#pragma unroll
    for (int dout = 0; dout < 16; ++dout) op[dout * (HO * HO) + p] = acc[dout];
  }
}

// ---------------- kernel 5: pass 1 — per-column (over j) online max/sum of exp(S)
__global__ __launch_bounds__(32)
void pass1_colstats(const float* __restrict__ qbuf, const float* __restrict__ kbuf,
                    float* __restrict__ colmax, float* __restrict__ colsum) {
  const int b  = blockIdx.y;
  const int i0 = blockIdx.x * 16;
  const int lane = threadIdx.x;
  const int li = lane & 15;
  const int hi = lane >> 4;

  const float* kb = kbuf + ((size_t)b * NPAD + i0) * 16;
  const v2f* krow = (const v2f*)(kb + li * 16);
  v2f Bt[4];                       // B = K^T chunks (K rows = e)
#pragma unroll
  for (int c = 0; c < 4; ++c) Bt[c] = krow[2 * c + hi];

  const float* qb = qbuf + (size_t)b * (JROWS * 16);
  float m_run = -INFINITY, z_run = 0.f;

  for (int jt = 0; jt < JTILES; ++jt) {
    const v2f* qrow = (const v2f*)(qb + (jt * 16 + li) * 16);
    v8f S = {};
#pragma unroll
    for (int c = 0; c < 4; ++c) {
      v2f A = qrow[2 * c + hi] * 0.25f;   // fold 1/sqrt(d)
      S = wmma_f32_4(A, Bt[c], S);
    }
    float tmax = S[0];
#pragma unroll
    for (int r = 1; r < 8; ++r) tmax = fmaxf(tmax, S[r]);
    tmax = fmaxf(tmax, __shfl_xor(tmax, 16, 32));
    const float m_new = fmaxf(m_run, tmax);
    float zt = 0.f;
#pragma unroll
    for (int r = 0; r < 8; ++r) zt += __expf(S[r] - m_new);
    zt += __shfl_xor(zt, 16, 32);
    z_run = z_run * __expf(m_run - m_new) + zt;
    m_run = m_new;
  }
  if (lane < 16) {
    colmax[(size_t)b * NPAD + i0 + lane] = m_run;
    colsum[(size_t)b * NPAD + i0 + lane] = z_run;
  }
}

// ---------------- kernel 6: pass 2 — O[j,:] = sum_i softmax_col(S)[j,i] * K[i,:]
__global__ __launch_bounds__(32)
void pass2_attend(const float* __restrict__ qbuf, const float* __restrict__ kbuf,
                  const float* __restrict__ colmax, const float* __restrict__ colsum,
                  float* __restrict__ outmid) {
  __shared__ float pt[16 * 16];    // P tile bounce buffer (C/D layout -> A layout)
  const int b  = blockIdx.y;
  const int j0 = blockIdx.x * 16;
  const int lane = threadIdx.x;
  const int li = lane & 15;
  const int hi = lane >> 4;

  const v2f* qrow = (const v2f*)(qbuf + ((size_t)b * JROWS + j0 + li) * 16);
  v2f Aq[4];
#pragma unroll
  for (int c = 0; c < 4; ++c) Aq[c] = qrow[2 * c + hi] * 0.25f;

  v8f O = {};
  for (int it = 0; it < NTILES; ++it) {
    const float* kt = kbuf + ((size_t)b * NPAD + it * 16) * 16;
    const v2f* krow = (const v2f*)(kt + li * 16);
    v8f S = {};
#pragma unroll
    for (int c = 0; c < 4; ++c)
      S = wmma_f32_4(Aq[c], krow[2 * c + hi], S);

    const float m  = colmax[(size_t)b * NPAD + it * 16 + li];
    const float rz = 1.f / colsum[(size_t)b * NPAD + it * 16 + li];
#pragma unroll
    for (int r = 0; r < 8; ++r)
      pt[(r + 8 * hi) * 16 + li] = __expf(S[r] - m) * rz;
    __syncthreads();

    const v2f* prow = (const v2f*)(pt + li * 16);
#pragma unroll
    for (int c = 0; c < 4; ++c) {
      const int e0 = 4 * c + 2 * hi;          // contraction (i) chunk
      v2f Ap = prow[2 * c + hi];              // A = P[16 j x 4 i]
      v2f Bk;                                 // B = K[4 i x 16 e]
      Bk.x = kt[e0 * 16 + li];
      Bk.y = kt[(e0 + 1) * 16 + li];
      O = wmma_f32_4(Ap, Bk, O);
    }
    __syncthreads();
  }
#pragma unroll
  for (int r = 0; r < 8; ++r)
    outmid[(size_t)b * (JROWS * 16) + (j0 + r + 8 * hi) * 16 + li] = O[r];
}

// ---------------- kernel 7: final grouped conv + LayerNorm -> d_out
__global__ __launch_bounds__(64)
void conv_out_ln_kernel(const float* __restrict__ outmid,
                        const float* __restrict__ next_w,
                        const float* __restrict__ gamma,
                        const float* __restrict__ beta,
                        float* __restrict__ out) {
  const int bn = blockIdx.x;
  const int b = bn >> 5, n = bn & 31;
  __shared__ float sIn[HO * HO * 16];       // flat [din][s]
  __shared__ float sW[16 * 16 * 9];
  const float* inp = outmid + (size_t)b * (JROWS * 16) + (size_t)(n * 16) * (HO * HO);
  const float* wp  = next_w + (size_t)(n * 16) * 144;
  for (int t = threadIdx.x; t < HO * HO * 16; t += 64) sIn[t] = inp[t];
  for (int t = threadIdx.x; t < 16 * 144;     t += 64) sW[t]  = wp[t];
  __syncthreads();
  const int p = threadIdx.x;
  if (p < HO * HO) {
    const int y = p / HO, x = p % HO;
    float acc[16];
#pragma unroll
    for (int d = 0; d < 16; ++d) acc[d] = 0.f;
    for (int ky = 0; ky < 3; ++ky) {
      const int yy = y + ky - 1;
      if (yy < 0 || yy >= HO) continue;
      for (int kx = 0; kx < 3; ++kx) {
        const int xx = x + kx - 1;
        if (xx < 0 || xx >= HO) continue;
        const int sp = yy * HO + xx;
        for (int din = 0; din < 16; ++din) {
          const float v = sIn[din * (HO * HO) + sp];
          const float* wv = &sW[din * 9 + ky * 3 + kx];
#pragma unroll
          for (int dout = 0; dout < 16; ++dout)
            acc[dout] += v * wv[dout * 144];
        }
      }
    }
    float mu = 0.f;
#pragma unroll
    for (int e = 0; e < 16; ++e) mu += acc[e];
    mu *= (1.f / 16.f);
    float var = 0.f;
#pragma unroll
    for (int e = 0; e < 16; ++e) { const float d = acc[e] - mu; var += d * d; }
    var *= (1.f / 16.f);
    const float inv = rsqrtf(var + LN_EPS);
    float* op = out + ((size_t)bn * (HO * HO) + p) * 16;
#pragma unroll
    for (int e = 0; e < 16; ++e)
      op[e] = (acc[e] - mu) * inv * gamma[e] + beta[e];
  }
}

// ---------------- launch ----------------
extern "C" void kernel_launch(void* const* d_in, const int* in_sizes, int n_in,
                              void* d_out, int out_size, void* d_ws, size_t ws_size,
                              hipStream_t stream) {
  (void)in_sizes; (void)n_in; (void)out_size; (void)ws_size;
  const float* cur_pose  = (const float*)d_in[0];
  const float* next_pose = (const float*)d_in[1];
  const float* cur_w     = (const float*)d_in[2];
  const float* next_w    = (const float*)d_in[3];
  const float* E_proj    = (const float*)d_in[4];
  const float* rel       = (const float*)d_in[5];
  const float* ln_gamma  = (const float*)d_in[6];
  const float* ln_beta   = (const float*)d_in[7];
  float* outp = (float*)d_out;

  float* ws     = (float*)d_ws;
  float* cpconv = ws;                                  // B*512*196
  float* kbuf   = cpconv + (size_t)BB * 512 * 196;     // B*NPAD*16
  float* qbuf   = kbuf   + (size_t)BB * NPAD * 16;     // B*1568*16
  float* cmax   = qbuf   + (size_t)BB * JROWS * 16;    // B*NPAD
  float* csum   = cmax   + (size_t)BB * NPAD;          // B*NPAD
  float* omid   = csum   + (size_t)BB * NPAD;          // B*1568*16

  conv_current_kernel<<<BB * IN_N, 256, 0, stream>>>(cur_pose, cur_w, cpconv);
  build_k_kernel<<<BB * IN_N, 256, 0, stream>>>(cpconv, E_proj, kbuf);
  fill_rel_kernel<<<BB, 256, 0, stream>>>(rel, kbuf);
  conv_q_kernel<<<BB * OUT_N, 64, 0, stream>>>(next_pose, next_w, qbuf);
  pass1_colstats<<<dim3(NTILES, BB), 32, 0, stream>>>(qbuf, kbuf, cmax, csum);
  pass2_attend<<<dim3(JTILES, BB), 32, 0, stream>>>(qbuf, kbuf, cmax, csum, omid);
  conv_out_ln_kernel<<<BB * OUT_N, 64, 0, stream>>>(omid, next_w, ln_gamma, ln_beta, outp);
}